// Net_60576218743439
// MI455X (gfx1250) — compile-verified
//
#include <hip/hip_runtime.h>
#include <hip/hip_bf16.h>
#include <math.h>

// ---------------------------------------------------------------------------
// GAT 3-layer forward on gfx1250 (MI455X).
//   layer(x, W, a_s, a_d, b):
//     h = x @ W^T                       (WMMA f32 16x16x4 GEMM)
//     als[n,hd] = dot(h[n,hd,:], a_s)   (per-node alpha)
//     ald[n,hd] = dot(h[n,hd,:], a_d)
//     segment softmax over dst (atomic max / atomic add)
//     out[dst] += alpha * h[src]        (atomic f32 scatter) + bias, relu
// ---------------------------------------------------------------------------

typedef __attribute__((ext_vector_type(2))) float v2f;
typedef __attribute__((ext_vector_type(8))) float v8f;

#define WAVES_PER_BLOCK 4
#define NEG_SLOPE 0.2f

// ---------------------------------------------------------------------------
// h = x @ W^T, x:[N,64], W:[64,64] (row = out feature), h:[N,64]
// One wave32 per 16-row tile; 4 accumulators cover all 64 output features.
// A fragment (16x4 f32): lanes 0-15 -> M=lane, K={k0,k0+1};
//                        lanes 16-31 -> M=lane-16, K={k0+2,k0+3}.
// B fragment (4x16 f32): same K split, N = lane&15 within each n-tile.
// ---------------------------------------------------------------------------
__global__ __launch_bounds__(128) void gat_gemm_wmma(
    const float* __restrict__ x, const float* __restrict__ W,
    float* __restrict__ h, int N) {
  const int lane  = threadIdx.x & 31;
  const int wave  = threadIdx.x >> 5;
  const int tile  = blockIdx.x * WAVES_PER_BLOCK + wave;
  const int rowBase = tile * 16;
  if (rowBase >= N) return;            // wave-uniform: EXEC stays all-ones
  const int mrow  = lane & 15;
  const int khalf = (lane >> 4) << 1;  // 0 or 2
  int row = rowBase + mrow;
  if (row > N - 1) row = N - 1;        // clamp keeps all lanes active
  const float* xr = x + (size_t)row * 64;
  __builtin_prefetch(xr, 0, 1);        // global_prefetch_b8: pull row into WGP$

  v8f acc[4] = {};
  for (int k0 = 0; k0 < 64; k0 += 4) {
    const int kk = k0 + khalf;
    v2f a = *(const v2f*)(xr + kk);
#pragma unroll
    for (int nt = 0; nt < 4; ++nt) {
      const int ncol = nt * 16 + mrow;            // output feature index
      v2f b = *(const v2f*)(W + (size_t)ncol * 64 + kk);  // B[k,n] = W[n,k]
      acc[nt] = __builtin_amdgcn_wmma_f32_16x16x4_f32(
          false, a, false, b, (short)0, acc[nt], false, false);
    }
  }

  // D layout: vgpr r, lanes 0-15 -> M=r; lanes 16-31 -> M=r+8; N = lane&15
  const int rhalf = (lane >> 4) * 8;
#pragma unroll
  for (int nt = 0; nt < 4; ++nt) {
#pragma unroll
    for (int r = 0; r < 8; ++r) {
      const int rr = rowBase + rhalf + r;
      if (rr < N) h[(size_t)rr * 64 + nt * 16 + mrow] = acc[nt][r];
    }
  }
}

// ---------------------------------------------------------------------------
// Per-(node, head) attention logits: als = <h, a_src>, ald = <h, a_dst>
// ---------------------------------------------------------------------------
__global__ void gat_alpha(const float* __restrict__ h,
                          const float* __restrict__ a_src,
                          const float* __restrict__ a_dst,
                          float* __restrict__ als, float* __restrict__ ald,
                          int N) {
  const int g = blockIdx.x * blockDim.x + threadIdx.x;
  if (g >= N * 8) return;
  const int n = g >> 3, hd = g & 7;
  const float* hp  = h + (size_t)n * 64 + hd * 8;
  const float* sp  = a_src + hd * 8;
  const float* dp  = a_dst + hd * 8;
  float s = 0.f, d = 0.f;
#pragma unroll
  for (int c = 0; c < 8; ++c) { const float v = hp[c]; s += v * sp[c]; d += v * dp[c]; }
  als[g] = s;
  ald[g] = d;
}

__global__ void gat_init_out(float* __restrict__ out,
                             const float* __restrict__ bias, long long n64) {
  const long long g = (long long)blockIdx.x * blockDim.x + threadIdx.x;
  if (g < n64) out[g] = bias[g & 63];
}

__global__ void gat_init_stats(float* __restrict__ mx, float* __restrict__ dn,
                               int n8) {
  const int g = blockIdx.x * blockDim.x + threadIdx.x;
  if (g >= n8) return;
  mx[g] = __int_as_float(0xFF800000);  // -inf
  dn[g] = 0.f;
}

__device__ __forceinline__ float leaky(float v) {
  return v > 0.f ? v : NEG_SLOPE * v;
}

// IEEE-ordered float atomic max via int punning:
//  non-negative patterns order as signed ints, negative as reversed unsigned.
__device__ __forceinline__ void atomicMaxFloat(float* addr, float v) {
  if (v >= 0.f) atomicMax((int*)addr, __float_as_int(v));
  else          atomicMin((unsigned int*)addr, __float_as_uint(v));
}

// Pass 1: segment max of e = leaky(als[src] + ald[dst]) over dst
__global__ void gat_edge_max(const int* __restrict__ ei,
                             const float* __restrict__ als,
                             const float* __restrict__ ald,
                             float* __restrict__ mx, int E, int N) {
  const long long g = (long long)blockIdx.x * blockDim.x + threadIdx.x;
  const long long ET = (long long)E + N;
  if (g >= ET) return;
  const int s = (g < E) ? ei[g]     : (int)(g - E);
  const int d = (g < E) ? ei[E + g] : (int)(g - E);
  const float* sp = als + (size_t)s * 8;
  const float* dp = ald + (size_t)d * 8;
  float* mp = mx + (size_t)d * 8;
#pragma unroll
  for (int hd = 0; hd < 8; ++hd)
    atomicMaxFloat(&mp[hd], leaky(sp[hd] + dp[hd]));
}

// Pass 2: denom[dst] += exp(e - max[dst])
__global__ void gat_edge_sum(const int* __restrict__ ei,
                             const float* __restrict__ als,
                             const float* __restrict__ ald,
                             const float* __restrict__ mx,
                             float* __restrict__ dn, int E, int N) {
  const long long g = (long long)blockIdx.x * blockDim.x + threadIdx.x;
  const long long ET = (long long)E + N;
  if (g >= ET) return;
  const int s = (g < E) ? ei[g]     : (int)(g - E);
  const int d = (g < E) ? ei[E + g] : (int)(g - E);
  const float* sp = als + (size_t)s * 8;
  const float* dp = ald + (size_t)d * 8;
  const float* mp = mx + (size_t)d * 8;
  float* np = dn + (size_t)d * 8;
#pragma unroll
  for (int hd = 0; hd < 8; ++hd)
    atomicAdd(&np[hd], __expf(leaky(sp[hd] + dp[hd]) - mp[hd]));
}

// Pass 3: out[dst, c] += alpha * h[src, c]; 64 threads per edge.
__global__ void gat_edge_scatter(const int* __restrict__ ei,
                                 const float* __restrict__ als,
                                 const float* __restrict__ ald,
                                 const float* __restrict__ mx,
                                 const float* __restrict__ dn,
                                 const float* __restrict__ h,
                                 float* __restrict__ out, int E, int N) {
  const long long g = (long long)blockIdx.x * blockDim.x + threadIdx.x;
  const long long ET = (long long)E + N;
  if (g >= ET * 64) return;
  const long long e = g >> 6;
  const int c  = (int)(g & 63);
  const int hd = c >> 3;
  const int s = (e < E) ? ei[e]     : (int)(e - E);
  const int d = (e < E) ? ei[E + e] : (int)(e - E);
  const float t = leaky(als[(size_t)s * 8 + hd] + ald[(size_t)d * 8 + hd]);
  const float w = __expf(t - mx[(size_t)d * 8 + hd]) / dn[(size_t)d * 8 + hd];
  atomicAdd(&out[(size_t)d * 64 + c], w * h[(size_t)s * 64 + c]);
}

__global__ void gat_relu(float* __restrict__ out, long long n64) {
  const long long g = (long long)blockIdx.x * blockDim.x + threadIdx.x;
  if (g < n64) { const float v = out[g]; out[g] = v > 0.f ? v : 0.f; }
}

// ---------------------------------------------------------------------------
static void run_layer(const float* xin, const float* W, const float* a_s,
                      const float* a_d, const float* bias, float* out,
                      int relu, float* h, float* als, float* ald, float* mx,
                      float* dn, const int* ei, int N, int E,
                      hipStream_t stream) {
  const int tiles = (N + 15) / 16;
  const unsigned gblocks = (unsigned)((tiles + WAVES_PER_BLOCK - 1) / WAVES_PER_BLOCK);
  gat_gemm_wmma<<<gblocks, 128, 0, stream>>>(xin, W, h, N);

  gat_alpha<<<(unsigned)((N * 8 + 255) / 256), 256, 0, stream>>>(h, a_s, a_d, als, ald, N);

  const long long n64 = (long long)N * 64;
  gat_init_out<<<(unsigned)((n64 + 255) / 256), 256, 0, stream>>>(out, bias, n64);
  gat_init_stats<<<(unsigned)((N * 8 + 255) / 256), 256, 0, stream>>>(mx, dn, N * 8);

  const long long ET = (long long)E + N;
  const unsigned eblocks = (unsigned)((ET + 255) / 256);
  gat_edge_max<<<eblocks, 256, 0, stream>>>(ei, als, ald, mx, E, N);
  gat_edge_sum<<<eblocks, 256, 0, stream>>>(ei, als, ald, mx, dn, E, N);

  const long long tot = ET * 64;
  gat_edge_scatter<<<(unsigned)((tot + 255) / 256), 256, 0, stream>>>(
      ei, als, ald, mx, dn, h, out, E, N);

  if (relu)
    gat_relu<<<(unsigned)((n64 + 255) / 256), 256, 0, stream>>>(out, n64);
}

extern "C" void kernel_launch(void* const* d_in, const int* in_sizes, int n_in,
                              void* d_out, int out_size, void* d_ws,
                              size_t ws_size, hipStream_t stream) {
  const float* x  = (const float*)d_in[0];
  const int*   ei = (const int*)d_in[1];
  const float* W1  = (const float*)d_in[2];
  const float* as1 = (const float*)d_in[3];
  const float* ad1 = (const float*)d_in[4];
  const float* b1  = (const float*)d_in[5];
  const float* W2  = (const float*)d_in[6];
  const float* as2 = (const float*)d_in[7];
  const float* ad2 = (const float*)d_in[8];
  const float* b2  = (const float*)d_in[9];
  const float* W3  = (const float*)d_in[10];
  const float* as3 = (const float*)d_in[11];
  const float* ad3 = (const float*)d_in[12];
  const float* b3  = (const float*)d_in[13];

  const int N = in_sizes[0] / 64;
  const int E = in_sizes[1] / 2;
  const long long N64 = (long long)N * 64;
  const long long N8  = (long long)N * 8;

  float* ws   = (float*)d_ws;
  float* h    = ws;                 // [N,64]
  float* outA = h + N64;            // [N,64]
  float* outB = outA + N64;         // [N,64]
  float* als  = outB + N64;         // [N,8]
  float* ald  = als + N8;           // [N,8]
  float* mx   = ald + N8;           // [N,8]
  float* dn   = mx + N8;            // [N,8]

  float* y = (float*)d_out;

  // conv1 + relu
  run_layer(x,    W1, as1, ad1, b1, outA, 1, h, als, ald, mx, dn, ei, N, E, stream);
  // conv3 + relu
  run_layer(outA, W3, as3, ad3, b3, outB, 1, h, als, ald, mx, dn, ei, N, E, stream);
  // conv2 (embedding, no relu) -> d_out
  run_layer(outB, W2, as2, ad2, b2, y,    0, h, als, ald, mx, dn, ei, N, E, stream);
}